// FastFood_Layer_67439576482217
// MI455X (gfx1250) — compile-verified
//
#include <hip/hip_runtime.h>
#include <stdint.h>

// FastFood layer for MI455X (gfx1250, wave32).
// D=1024, R=4, rows M = 16384. out[row][r*1024+j].
// FWHT_1024 = (H4 (x) H16) (x) H16 done with V_WMMA_F32_16X16X4_F32 tiles
// (exact f32 math; H entries are +-1). Memory-bound: ~320MB @ 23.3TB/s.

#define FF_D 1024
#define FF_R 4
#define WAVES 4
#define ROWS_PER_WAVE 4
#define BLOCK_ROWS (WAVES * ROWS_PER_WAVE)

typedef __attribute__((ext_vector_type(2))) float v2f;
typedef __attribute__((ext_vector_type(8))) float v8f;
typedef __attribute__((ext_vector_type(4))) int v4i;

#define FF_GAS __attribute__((address_space(1)))
#define FF_LAS __attribute__((address_space(3)))

#if __has_builtin(__builtin_amdgcn_wmma_f32_16x16x4_f32)
#define FF_HAVE_WMMA 1
#else
#define FF_HAVE_WMMA 0
#endif

#if __has_builtin(__builtin_amdgcn_global_load_async_to_lds_b128) && \
    __has_builtin(__builtin_amdgcn_global_store_async_from_lds_b128)
#define FF_HAVE_ASYNC 1
#else
#define FF_HAVE_ASYNC 0
#endif

static __device__ __forceinline__ void ff_wait_async0() {
#if __has_builtin(__builtin_amdgcn_s_wait_asynccnt)
  __builtin_amdgcn_s_wait_asynccnt(0);
#else
  asm volatile("s_wait_asynccnt 0" ::: "memory");
#endif
}

static __device__ __forceinline__ void ff_wait_ds0() {
#if __has_builtin(__builtin_amdgcn_s_wait_dscnt)
  __builtin_amdgcn_s_wait_dscnt(0);
#else
  asm volatile("s_wait_dscnt 0" ::: "memory");
#endif
}

// 16B per lane, global -> LDS (async if available)
static __device__ __forceinline__ void ff_g2l_16B(const float* g, float* l) {
#if FF_HAVE_ASYNC
  __builtin_amdgcn_global_load_async_to_lds_b128(
      (FF_GAS v4i*)g, (FF_LAS v4i*)l, 0, 0);
#else
  float4 v = *(const float4*)g;
  *(float4*)l = v;
#endif
}

// 16B per lane, LDS -> global (async if available)
static __device__ __forceinline__ void ff_l2g_16B(float* g, const float* l) {
#if FF_HAVE_ASYNC
  __builtin_amdgcn_global_store_async_from_lds_b128(
      (FF_GAS v4i*)g, (FF_LAS v4i*)l, 0, 0);
#else
  float4 v = *(const float4*)l;
  *(float4*)g = v;
#endif
}

#if FF_HAVE_WMMA
// in:  linear 1024 floats viewed as X[u][v], u in [0,64), v in [0,16)
// mul: optional elementwise multiplier (same linear layout), may be nullptr
// tmp_out: scratch AND final destination (linear FWHT result), != in
// Hc: H16 +-1 constants in the shared A/B f32 WMMA lane layout (H16 symmetric)
static __device__ __forceinline__ void ff_fwht1024_wmma(
    const float* __restrict__ in, const float* __restrict__ mul,
    float* __restrict__ tmp_out, int lane, const v2f Hc[4]) {
  const int hi = lane >> 4;   // 0 or 1
  const int lo = lane & 15;

  // ---- step 1: Z = X * H16  (right-multiply over v), 4 tiles of 16 rows
#pragma unroll
  for (int m = 0; m < 4; ++m) {
    v2f a[4];
#pragma unroll
    for (int kk = 0; kk < 4; ++kk) {
      const int c = 4 * kk + 2 * hi;               // K column pair
      const int idx = (16 * m + lo) * 16 + c;      // A[M=lo][K=c], row u=16m+lo
      v2f xa = *(const v2f*)(in + idx);
      if (mul) {
        v2f mb = *(const v2f*)(mul + idx);
        xa *= mb;
      }
      a[kk] = xa;
    }
    v8f acc = {0.f, 0.f, 0.f, 0.f, 0.f, 0.f, 0.f, 0.f};
#pragma unroll
    for (int kk = 0; kk < 4; ++kk)
      acc = __builtin_amdgcn_wmma_f32_16x16x4_f32(
          false, a[kk], false, Hc[kk], (short)0, acc, false, false);
    // D[M][N]: M = g + 8*hi, N = lo  -> Z[16m+M][N]
#pragma unroll
    for (int g = 0; g < 8; ++g)
      tmp_out[(16 * m + g + 8 * hi) * 16 + lo] = acc[g];
  }
  __builtin_amdgcn_wave_barrier();

  // ---- step 2: W_p = H16 * Z_p (left-multiply over q), p in [0,4)
  v8f w[4];
#pragma unroll
  for (int p = 0; p < 4; ++p) {
    v8f acc = {0.f, 0.f, 0.f, 0.f, 0.f, 0.f, 0.f, 0.f};
#pragma unroll
    for (int kk = 0; kk < 4; ++kk) {
      const int c = 4 * kk + 2 * hi;               // K row pair of B
      v2f b;
      b.x = tmp_out[(p * 16 + c) * 16 + lo];       // B[K=c][N=lo]
      b.y = tmp_out[(p * 16 + c + 1) * 16 + lo];
      acc = __builtin_amdgcn_wmma_f32_16x16x4_f32(
          false, Hc[kk], false, b, (short)0, acc, false, false);
    }
    w[p] = acc;
  }
  __builtin_amdgcn_wave_barrier();

  // ---- step 3: H4 across p in VALU
  v8f s0 = w[0] + w[1], d0 = w[0] - w[1];
  v8f s1 = w[2] + w[3], d1 = w[2] - w[3];
  v8f y[4] = {s0 + s1, d0 + d1, s0 - s1, d0 - d1};
  // y[kp] element (M=g+8*hi, N=lo) is FWHT index kp*256 + M*16 + N
#pragma unroll
  for (int p = 0; p < 4; ++p)
#pragma unroll
    for (int g = 0; g < 8; ++g)
      tmp_out[p * 256 + (g + 8 * hi) * 16 + lo] = y[p][g];
  __builtin_amdgcn_wave_barrier();
}
#else
// VALU butterfly fallback (only if the f32 WMMA builtin is absent)
static __device__ __forceinline__ void ff_fwht1024_valu(
    const float* __restrict__ in, const float* __restrict__ mul,
    float* __restrict__ out, int lane) {
#pragma unroll
  for (int blk = 0; blk < 8; ++blk) {
    const int j = lane * 32 + blk * 4;
    float4 v = *(const float4*)(in + j);
    if (mul) {
      const float4 m4 = *(const float4*)(mul + j);
      v.x *= m4.x; v.y *= m4.y; v.z *= m4.z; v.w *= m4.w;
    }
    *(float4*)(out + j) = v;
  }
  __builtin_amdgcn_wave_barrier();
  for (int h = 1; h < 1024; h <<= 1) {
#pragma unroll
    for (int t = 0; t < 16; ++t) {
      const int tt = lane + 32 * t;                       // 512 butterflies
      const int i0 = ((tt & ~(h - 1)) << 1) | (tt & (h - 1));
      const float a = out[i0];
      const float b = out[i0 + h];
      out[i0] = a + b;
      out[i0 + h] = a - b;
    }
    __builtin_amdgcn_wave_barrier();
  }
}
#endif

__global__ __launch_bounds__(WAVES * 32) void ff_kernel(
    const float* __restrict__ x, const float* __restrict__ B,
    const float* __restrict__ G, const float* __restrict__ S,
    const int* __restrict__ P, float* __restrict__ out) {
  __shared__ __align__(16) float Bs[FF_D];
  __shared__ __align__(16) float Gs[FF_D];
  __shared__ __align__(16) float Ss[FF_D];
  __shared__ __align__(16) int Ps[FF_D];
  __shared__ __align__(16) float bufA_s[WAVES][FF_D];
  __shared__ __align__(16) float bufB_s[WAVES][FF_D];

  const int r = blockIdx.y;
  const int tid = threadIdx.x;
  const int wid = tid >> 5;
  const int lane = tid & 31;

  // stage per-r params once per block (1/sqrt(1024) folded into S)
  for (int i = tid; i < FF_D; i += WAVES * 32) {
    Bs[i] = B[r * FF_D + i];
    Gs[i] = G[r * FF_D + i];
    Ss[i] = S[r * FF_D + i] * 0.03125f;
    Ps[i] = P[r * FF_D + i];
  }
  __syncthreads();

  const int hi = lane >> 4;
  const int lo = lane & 15;
  v2f Hc[4];
#pragma unroll
  for (int kk = 0; kk < 4; ++kk) {
    const int c = 4 * kk + 2 * hi;
    Hc[kk].x = (__popc(c & lo) & 1) ? -1.0f : 1.0f;
    Hc[kk].y = (__popc((c + 1) & lo) & 1) ? -1.0f : 1.0f;
  }

  float* bufA = &bufA_s[wid][0];
  float* bufB = &bufB_s[wid][0];

  const long long rowBase =
      ((long long)blockIdx.x * WAVES + wid) * ROWS_PER_WAVE;

  for (int it = 0; it < ROWS_PER_WAVE; ++it) {
    const long long row = rowBase + it;
    const float* xrow = x + row * FF_D;

    // drain our DS pipe, then async-load the row; the asynccnt wait below also
    // covers last iteration's async store out of bufB before we rewrite it.
    ff_wait_ds0();
#pragma unroll
    for (int is = 0; is < 8; ++is)
      ff_g2l_16B(xrow + is * 128 + lane * 4, bufA + is * 128 + lane * 4);
    ff_wait_async0();
    __builtin_amdgcn_wave_barrier();

    // h1 = FWHT(x * B)  -> bufB (linear)
#if FF_HAVE_WMMA
    ff_fwht1024_wmma(bufA, Bs, bufB, lane, Hc);
#else
    ff_fwht1024_valu(bufA, Bs, bufB, lane);
#endif

    // t[j] = h1[P[j]] * G[j] -> bufA (vectorized index/coef loads, scalar gather)
#pragma unroll
    for (int blk = 0; blk < 8; ++blk) {
      const int j = lane * 32 + blk * 4;
      const int4 p4 = *(const int4*)(Ps + j);
      const float4 g4 = *(const float4*)(Gs + j);
      float4 t;
      t.x = bufB[p4.x] * g4.x;
      t.y = bufB[p4.y] * g4.y;
      t.z = bufB[p4.z] * g4.z;
      t.w = bufB[p4.w] * g4.w;
      *(float4*)(bufA + j) = t;
    }
    __builtin_amdgcn_wave_barrier();

    // h2 = FWHT(t) -> bufB (linear)
#if FF_HAVE_WMMA
    ff_fwht1024_wmma(bufA, nullptr, bufB, lane, Hc);
#else
    ff_fwht1024_valu(bufA, nullptr, bufB, lane);
#endif

    // scale by S/sqrt(D) in place (keeps the global store 16B coalesced)
#pragma unroll
    for (int blk = 0; blk < 8; ++blk) {
      const int j = lane * 32 + blk * 4;
      float4 v = *(float4*)(bufB + j);
      const float4 s4 = *(const float4*)(Ss + j);
      v.x *= s4.x; v.y *= s4.y; v.z *= s4.z; v.w *= s4.w;
      *(float4*)(bufB + j) = v;
    }
    __builtin_amdgcn_wave_barrier();
    ff_wait_ds0();  // scaled data must be in LDS before the async engine reads it

    float* orow = out + row * (long long)(FF_R * FF_D) + r * FF_D;
#pragma unroll
    for (int is = 0; is < 8; ++is)
      ff_l2g_16B(orow + is * 128 + lane * 4, bufB + is * 128 + lane * 4);
  }
  ff_wait_async0();
}

extern "C" void kernel_launch(void* const* d_in, const int* in_sizes, int n_in,
                              void* d_out, int out_size, void* d_ws,
                              size_t ws_size, hipStream_t stream) {
  (void)n_in; (void)out_size; (void)d_ws; (void)ws_size;
  const float* x = (const float*)d_in[0];
  const float* B = (const float*)d_in[1];
  const float* G = (const float*)d_in[2];
  const float* S = (const float*)d_in[3];
  const int*   P = (const int*)d_in[4];
  float* out = (float*)d_out;

  const int Mrows = in_sizes[0] / FF_D;  // 16384
  dim3 grid((unsigned)(Mrows / BLOCK_ROWS), FF_R);
  dim3 block(WAVES * 32);
  hipLaunchKernelGGL(ff_kernel, grid, block, 0, stream, x, B, G, S, P, out);
}